// SparseToDense_49022756717277
// MI455X (gfx1250) — compile-verified
//
#include <hip/hip_runtime.h>

// ---------------------------------------------------------------------------
// SparseToDense: scatter-add 262144 x 64 f32 features into a dense
// [B=8, C=64, 64^3] channels-first grid (512 MiB), after zero-filling it.
// Pure data movement; roofline ~0.6-1.2 GB HBM traffic -> ~30-55 us @23.3TB/s.
//   out offset = (b << 24) | (c << 18) | (idx & 0x3FFFF)
// ---------------------------------------------------------------------------
#define LOG2_C   6      // C = 64
#define LOG2_S3  18     // S^3 = 262144

// Types for gfx1250 builtins (param 1 of the async-LDS builtin is int4* per
// the round-2 diagnostic; AS1 = global, AS3 = LDS).
typedef __attribute__((ext_vector_type(4))) int      v4i;
typedef __attribute__((address_space(1))) v4i*       g_v4i_p;
typedef __attribute__((address_space(3))) v4i*       l_v4i_p;
typedef __attribute__((address_space(1))) float*     g_float_p;

#if __has_builtin(__builtin_amdgcn_global_load_async_to_lds_b128)
#define HAVE_ASYNC_LDS 1
#else
#define HAVE_ASYNC_LDS 0
#endif

__device__ __forceinline__ void wait_asynccnt0() {
#if __has_builtin(__builtin_amdgcn_s_wait_asynccnt)
    __builtin_amdgcn_s_wait_asynccnt(0);
#else
    asm volatile("s_wait_asynccnt 0" ::: "memory");
#endif
}

__device__ __forceinline__ void atomic_add_f32_global(float* p, float v) {
#if __has_builtin(__builtin_amdgcn_global_atomic_fadd_f32)
    // Native no-return GLOBAL_ATOMIC_ADD_F32 (STOREcnt-tracked).
    (void)__builtin_amdgcn_global_atomic_fadd_f32((g_float_p)p, v);
#else
    (void)__hip_atomic_fetch_add(p, v, __ATOMIC_RELAXED,
                                 __HIP_MEMORY_SCOPE_AGENT);
#endif
}

// ---------------------------------------------------------------------------
// Never-launched probe kernel, FIRST in the file so its disassembly appears
// at the top of the feedback snippet: shows exactly which instruction the
// f32 scatter atomic lowers to (want: global_atomic_add_f32, no CAS loop).
// ---------------------------------------------------------------------------
__global__ void zz_probe_atomic_fadd(float* __restrict__ p, float v) {
    atomic_add_f32_global(p + threadIdx.x, v);
}

// ---------------------------------------------------------------------------
// Kernel A: scatter-add. One thread per (site, channel). The block's 1 KB
// feature chunk is staged Global->LDS on the CDNA5 async path
// (GLOBAL_LOAD_ASYNC_TO_LDS_B128, ASYNCcnt-tracked), then each thread reads
// its float from LDS and issues a native f32 global atomic add. Atomics are
// required: random idx over 2M cells with 262K draws collides, and the
// reference does `.add`.
// ---------------------------------------------------------------------------
__global__ __launch_bounds__(256) void stod_scatter_add(
        const float* __restrict__ feat,
        const long long* __restrict__ idx,
        float* __restrict__ out,
        unsigned total /* = N*C, multiple of 256 */) {
    const unsigned t = blockIdx.x * 256u + threadIdx.x;
    if (t >= total) return;

    const unsigned n = t >> LOG2_C;                  // active-site id
    const unsigned c = t & ((1u << LOG2_C) - 1u);    // channel

    const long long cell = idx[n];                   // flat (b,z,y,x)
    const unsigned long long ucell = (unsigned long long)cell;
    const unsigned long long b  = ucell >> LOG2_S3;
    const unsigned long long sp = ucell & ((1ull << LOG2_S3) - 1ull);
    const unsigned long long o  = (b << (LOG2_C + LOG2_S3))
                                | ((unsigned long long)c << LOG2_S3)
                                | sp;

#if HAVE_ASYNC_LDS
    __shared__ float stage[256];
    if (threadIdx.x < 64u) {
        // 64 lanes x 16B = the block's whole 1 KB feature chunk, one shot.
        float* gsrc = const_cast<float*>(feat) +
                      (size_t)blockIdx.x * 256u + threadIdx.x * 4u;
        __builtin_amdgcn_global_load_async_to_lds_b128(
            (g_v4i_p)gsrc,
            (l_v4i_p)&stage[threadIdx.x * 4u],
            /*imm offset*/ 0, /*cpol*/ 0);
    }
    wait_asynccnt0();     // issuing waves drain ASYNCcnt; no-op for the rest
    __syncthreads();      // LDS now holds the chunk for all 8 waves
    const float v = stage[threadIdx.x];
#else
    const float v = __builtin_nontemporal_load(&feat[t]);
#endif

    atomic_add_f32_global(&out[o], v);
}

// ---------------------------------------------------------------------------
// Kernel B: zero-fill the dense output, 16 bytes/thread -> global_store_b128.
// Regular (RT) temporal stores on purpose: the tail of the fill stays dirty in
// the 192 MB L2, so a fraction of the random scatter atomics hit zeros in L2
// instead of forcing an HBM read-modify-write.
// ---------------------------------------------------------------------------
__global__ __launch_bounds__(256) void stod_fill_zero(float4* __restrict__ out,
                                                      unsigned n4) {
    unsigned i = blockIdx.x * 256u + threadIdx.x;
    if (i < n4) {
        float4 z;
        z.x = 0.0f; z.y = 0.0f; z.z = 0.0f; z.w = 0.0f;
        out[i] = z;
    }
}

extern "C" void kernel_launch(void* const* d_in, const int* in_sizes, int n_in,
                              void* d_out, int out_size, void* d_ws, size_t ws_size,
                              hipStream_t stream) {
    const float*     feat = (const float*)d_in[0];      // [N, 64] f32
    const long long* idx  = (const long long*)d_in[1];  // [N] int64
    float*           out  = (float*)d_out;              // [8, 64, 64, 64, 64] f32

    const unsigned total = (unsigned)in_sizes[0];       // N*C = 16,777,216
    const unsigned n4    = (unsigned)out_size >> 2;     // 134,217,728 / 4

    stod_fill_zero<<<(n4 + 255u) / 256u, 256, 0, stream>>>(
        (float4*)out, n4);

    stod_scatter_add<<<(total + 255u) / 256u, 256, 0, stream>>>(
        feat, idx, out, total);
}